// MultiTransformer_4870492914033
// MI455X (gfx1250) — compile-verified
//
#include <hip/hip_runtime.h>
#include <math.h>

#define N_NODES 50000
#define N_EDGES 400000
#define HCH 128          // H*C
#define EDIM 32
#define NHEAD 4

typedef __attribute__((ext_vector_type(2))) float v2f;
typedef __attribute__((ext_vector_type(8))) float v8f;

// ---------------------------------------------------------------------------
// C[M x 128] = A[M x K] @ W[K x 128] + bias[128]
// One wave computes one 16x16 tile via chained V_WMMA_F32_16X16X4_F32.
// Block = 256 threads = 8 waves, wave w owns output columns [16w, 16w+16).
// M must be a multiple of 16 (50000 and 400000 both are).
// A layout (16x4 f32): lanes 0-15 row M=l16 hold K=k0,k0+1 in v0,v1;
//                      lanes 16-31 row M=l16 hold K=k0+2,k0+3.
// B layout mirrored: lanes hold column N=l16, same K split.
// C/D: VGPR r -> row r (lanes 0-15) / row r+8 (lanes 16-31), col = l16.
// ---------------------------------------------------------------------------
template <int K>
__global__ __launch_bounds__(256) void gemm_wmma(const float* __restrict__ A,
                                                 const float* __restrict__ W,
                                                 const float* __restrict__ bias,
                                                 float* __restrict__ Out) {
  const int lane = threadIdx.x & 31;
  const int wv   = threadIdx.x >> 5;   // 0..7  column tile
  const int half = lane >> 4;          // 0 / 1
  const int l16  = lane & 15;
  const int row0 = blockIdx.x * 16;

  const float* arow = A + (size_t)(row0 + l16) * K + 2 * half;
  const float* wcol = W + 16 * wv + l16;
  const float  bb   = bias[16 * wv + l16];

  v8f c;
#pragma unroll
  for (int r = 0; r < 8; ++r) c[r] = bb;   // bias folded into accumulator

#pragma unroll
  for (int k0 = 0; k0 < K; k0 += 4) {
    float2 av = *(const float2*)(arow + k0);   // K=k0+2*half, k0+2*half+1
    v2f a, b;
    a.x = av.x;
    a.y = av.y;
    b.x = wcol[(size_t)(k0 + 2 * half + 0) * HCH];
    b.y = wcol[(size_t)(k0 + 2 * half + 1) * HCH];
    c = __builtin_amdgcn_wmma_f32_16x16x4_f32(false, a, false, b, (short)0, c,
                                              false, false);
  }

  float* orow = Out + (size_t)row0 * HCH + 16 * wv + l16;
#pragma unroll
  for (int r = 0; r < 8; ++r)
    orow[(size_t)(r + 8 * half) * HCH] = c[r];
}

// ---------------------------------------------------------------------------
__global__ void init_seg(float* __restrict__ amax, float* __restrict__ asum,
                         float* __restrict__ outm) {
  int i = blockIdx.x * blockDim.x + threadIdx.x;
  if (i < N_NODES * HCH) outm[i] = 0.0f;
  if (i < N_NODES * NHEAD) {
    amax[i] = -1.0e30f;
    asum[i] = 0.0f;
  }
}

__device__ inline void atomicMaxF(float* addr, float val) {
  unsigned int* ai  = (unsigned int*)addr;
  unsigned int  old = *ai;
  while (__uint_as_float(old) < val) {
    unsigned int assumed = old;
    old = atomicCAS(ai, assumed, __float_as_uint(val));
    if (old == assumed) break;
  }
}

// wave per edge: alpha[e][h] = (1/sqrt(C)) * sum_c q[dst]*(k[src]+e)
__global__ __launch_bounds__(256) void edge_alpha(
    const float* __restrict__ q, const float* __restrict__ k,
    const float* __restrict__ e, const int* __restrict__ src,
    const int* __restrict__ dst, float* __restrict__ alpha,
    float* __restrict__ amax) {
  int wid  = (blockIdx.x * blockDim.x + threadIdx.x) >> 5;
  int lane = threadIdx.x & 31;
  if (wid >= N_EDGES) return;
  int s = src[wid], d = dst[wid];

  float4 qv = *(const float4*)(q + (size_t)d * HCH + 4 * lane);
  float4 kv = *(const float4*)(k + (size_t)s * HCH + 4 * lane);
  float4 ev = *(const float4*)(e + (size_t)wid * HCH + 4 * lane);
  float p = qv.x * (kv.x + ev.x) + qv.y * (kv.y + ev.y) +
            qv.z * (kv.z + ev.z) + qv.w * (kv.w + ev.w);
  // reduce within each head group of 8 lanes (8 lanes * 4ch = C=32)
  p += __shfl_xor(p, 1);
  p += __shfl_xor(p, 2);
  p += __shfl_xor(p, 4);
  if ((lane & 7) == 0) {
    int   h = lane >> 3;
    float a = p * 0.17677669529663687f;  // 1/sqrt(32)
    alpha[(size_t)wid * NHEAD + h] = a;
    atomicMaxF(&amax[(size_t)d * NHEAD + h], a);
  }
}

// thread per (edge, head): aexp = exp(alpha - amax[dst]); asum += aexp
__global__ void edge_exp(const int* __restrict__ dst,
                         const float* __restrict__ amax,
                         float* __restrict__ alpha, float* __restrict__ asum) {
  int i = blockIdx.x * blockDim.x + threadIdx.x;
  if (i >= N_EDGES * NHEAD) return;
  int   eidx = i >> 2, h = i & 3;
  int   d  = dst[eidx];
  float ex = __expf(alpha[i] - amax[(size_t)d * NHEAD + h]);
  alpha[i] = ex;
  atomicAdd(&asum[(size_t)d * NHEAD + h], ex);
}

// wave per edge: out[dst] += (v[src]+e) * attn
__global__ __launch_bounds__(256) void edge_scatter(
    const float* __restrict__ v, const float* __restrict__ e,
    const float* __restrict__ aexp, const float* __restrict__ asum,
    const int* __restrict__ src, const int* __restrict__ dst,
    float* __restrict__ outm) {
  int wid  = (blockIdx.x * blockDim.x + threadIdx.x) >> 5;
  int lane = threadIdx.x & 31;
  if (wid >= N_EDGES) return;
  int s = src[wid], d = dst[wid];
  int h = lane >> 3;
  float attn = aexp[(size_t)wid * NHEAD + h] /
               (asum[(size_t)d * NHEAD + h] + 1e-16f);
  float4 vv = *(const float4*)(v + (size_t)s * HCH + 4 * lane);
  float4 ev = *(const float4*)(e + (size_t)wid * HCH + 4 * lane);
  float* base = outm + (size_t)d * HCH + 4 * lane;
  atomicAdd(base + 0, (vv.x + ev.x) * attn);
  atomicAdd(base + 1, (vv.y + ev.y) * attn);
  atomicAdd(base + 2, (vv.z + ev.z) * attn);
  atomicAdd(base + 3, (vv.w + ev.w) * attn);
}

// wave per node: beta gate + layernorm + leaky relu
__global__ __launch_bounds__(256) void node_final(
    const float* __restrict__ outm, const float* __restrict__ xr,
    const float* __restrict__ Wbeta, const float* __restrict__ lnw,
    const float* __restrict__ lnb, float* __restrict__ y) {
  int wid  = (blockIdx.x * blockDim.x + threadIdx.x) >> 5;
  int lane = threadIdx.x & 31;
  if (wid >= N_NODES) return;

  float4 o = *(const float4*)(outm + (size_t)wid * HCH + 4 * lane);
  float4 r = *(const float4*)(xr + (size_t)wid * HCH + 4 * lane);
  float4 w0 = *(const float4*)(Wbeta + 4 * lane);
  float4 w1 = *(const float4*)(Wbeta + HCH + 4 * lane);
  float4 w2 = *(const float4*)(Wbeta + 2 * HCH + 4 * lane);

  float p = o.x * w0.x + o.y * w0.y + o.z * w0.z + o.w * w0.w +
            r.x * w1.x + r.y * w1.y + r.z * w1.z + r.w * w1.w +
            (o.x - r.x) * w2.x + (o.y - r.y) * w2.y + (o.z - r.z) * w2.z +
            (o.w - r.w) * w2.w;
#pragma unroll
  for (int m = 1; m < 32; m <<= 1) p += __shfl_xor(p, m);
  float beta = 1.0f / (1.0f + __expf(-p));

  float4 h;
  h.x = beta * r.x + (1.0f - beta) * o.x;
  h.y = beta * r.y + (1.0f - beta) * o.y;
  h.z = beta * r.z + (1.0f - beta) * o.z;
  h.w = beta * r.w + (1.0f - beta) * o.w;

  float msum = h.x + h.y + h.z + h.w;
#pragma unroll
  for (int m = 1; m < 32; m <<= 1) msum += __shfl_xor(msum, m);
  float mu = msum * (1.0f / HCH);

  float4 hm;
  hm.x = h.x - mu; hm.y = h.y - mu; hm.z = h.z - mu; hm.w = h.w - mu;
  float vsum = hm.x * hm.x + hm.y * hm.y + hm.z * hm.z + hm.w * hm.w;
#pragma unroll
  for (int m = 1; m < 32; m <<= 1) vsum += __shfl_xor(vsum, m);
  float inv = rsqrtf(vsum * (1.0f / HCH) + 1e-5f);

  float4 lw = *(const float4*)(lnw + 4 * lane);
  float4 lb = *(const float4*)(lnb + 4 * lane);
  float4 z;
  z.x = hm.x * inv * lw.x + lb.x;
  z.y = hm.y * inv * lw.y + lb.y;
  z.z = hm.z * inv * lw.z + lb.z;
  z.w = hm.w * inv * lw.w + lb.w;
  z.x = z.x > 0.0f ? z.x : 0.01f * z.x;
  z.y = z.y > 0.0f ? z.y : 0.01f * z.y;
  z.z = z.z > 0.0f ? z.z : 0.01f * z.z;
  z.w = z.w > 0.0f ? z.w : 0.01f * z.w;
  *(float4*)(y + (size_t)wid * HCH + 4 * lane) = z;
}

// ---------------------------------------------------------------------------
extern "C" void kernel_launch(void* const* d_in, const int* in_sizes, int n_in,
                              void* d_out, int out_size, void* d_ws,
                              size_t ws_size, hipStream_t stream) {
  const float* x  = (const float*)d_in[0];
  const int*   ei = (const int*)d_in[1];  // [2, E]
  const float* ea = (const float*)d_in[2];
  const float* Wq = (const float*)d_in[3];
  const float* bq = (const float*)d_in[4];
  const float* Wk = (const float*)d_in[5];
  const float* bk = (const float*)d_in[6];
  const float* Wv = (const float*)d_in[7];
  const float* bv = (const float*)d_in[8];
  const float* We = (const float*)d_in[9];
  const float* be = (const float*)d_in[10];
  const float* Ws = (const float*)d_in[11];
  const float* bs = (const float*)d_in[12];
  const float* Wb = (const float*)d_in[13];
  const float* lw = (const float*)d_in[14];
  const float* lb = (const float*)d_in[15];

  const int* src = ei;
  const int* dst = ei + N_EDGES;

  float* p    = (float*)d_ws;
  float* q    = p; p += (size_t)N_NODES * HCH;
  float* kbuf = p; p += (size_t)N_NODES * HCH;
  float* vbuf = p; p += (size_t)N_NODES * HCH;
  float* xr   = p; p += (size_t)N_NODES * HCH;
  float* outm = p; p += (size_t)N_NODES * HCH;
  float* hbuf = p; p += (size_t)N_NODES * HCH;
  float* amax = p; p += (size_t)N_NODES * NHEAD;
  float* asum = p; p += (size_t)N_NODES * NHEAD;
  float* alph = p; p += (size_t)N_EDGES * NHEAD;
  float* ebuf = p; p += (size_t)N_EDGES * HCH;

  for (int l = 0; l < 2; ++l) {
    const float* xin  = l ? hbuf : x;
    float*       xout = l ? (float*)d_out : hbuf;
    const size_t wo = (size_t)l * HCH * HCH;
    const size_t bo = (size_t)l * HCH;

    gemm_wmma<HCH><<<N_NODES / 16, 256, 0, stream>>>(xin, Wq + wo, bq + bo, q);
    gemm_wmma<HCH><<<N_NODES / 16, 256, 0, stream>>>(xin, Wk + wo, bk + bo, kbuf);
    gemm_wmma<HCH><<<N_NODES / 16, 256, 0, stream>>>(xin, Wv + wo, bv + bo, vbuf);
    gemm_wmma<HCH><<<N_NODES / 16, 256, 0, stream>>>(xin, Ws + wo, bs + bo, xr);
    gemm_wmma<EDIM><<<N_EDGES / 16, 256, 0, stream>>>(
        ea, We + (size_t)l * EDIM * HCH, be + bo, ebuf);

    init_seg<<<(N_NODES * HCH + 255) / 256, 256, 0, stream>>>(amax, asum, outm);
    edge_alpha<<<(N_EDGES * 32) / 256, 256, 0, stream>>>(q, kbuf, ebuf, src,
                                                         dst, alph, amax);
    edge_exp<<<(N_EDGES * NHEAD + 255) / 256, 256, 0, stream>>>(dst, amax,
                                                                alph, asum);
    edge_scatter<<<(N_EDGES * 32) / 256, 256, 0, stream>>>(vbuf, ebuf, alph,
                                                           asum, src, dst,
                                                           outm);
    node_final<<<(N_NODES * 32) / 256, 256, 0, stream>>>(
        outm, xr, Wb + (size_t)l * 3 * HCH, lw + bo, lb + bo, xout);
  }
}